// SpikingProprioception_59777354826372
// MI455X (gfx1250) — compile-verified
//
#include <hip/hip_runtime.h>
#include <stdint.h>

#define NTHREADS 256
#define N_STEPS  10
#define NN       8

typedef float v2f __attribute__((ext_vector_type(2)));

// ---------------------------------------------------------------------------
// CDNA5 async global->LDS copy (VGLOBAL GLOBAL_LOAD_ASYNC_TO_LDS_B128,
// tracked by ASYNCcnt).  lds_byte_off is the per-lane LDS byte address
// (low 32 bits of a flat shared pointer == LDS offset per ISA 10.2).
// ---------------------------------------------------------------------------
__device__ __forceinline__ void async_ld16(uint32_t lds_byte_off, const float* g) {
    asm volatile("global_load_async_to_lds_b128 %0, %1, off"
                 :: "v"(lds_byte_off), "v"(g)
                 : "memory");
}

__device__ __forceinline__ void wait_async_le2() {
    asm volatile("s_wait_asynccnt 2" ::: "memory");
}
__device__ __forceinline__ void wait_async_0() {
    asm volatile("s_wait_asynccnt 0" ::: "memory");
}

__global__ __launch_bounds__(NTHREADS)
void fish_izhikevich_kernel(const float* __restrict__ fish_x,
                            const float* __restrict__ fish_y,
                            const float* __restrict__ speed,
                            const float* __restrict__ heading,
                            const float* __restrict__ prev_x,
                            const float* __restrict__ prev_y,
                            const float* __restrict__ prev_heading,
                            const float* __restrict__ pred_speed,
                            const float* __restrict__ pred_hd,
                            const float* __restrict__ v0,
                            const float* __restrict__ u0,
                            const float* __restrict__ noise,
                            float* __restrict__ out,
                            int n)
{
    // [buffer][chunk][lane][4 floats]: lane t's 16B chunk at byte t*16 of a
    // 4KB contiguous region -> minimum bank phases for wave32 ds_load_b128.
    __shared__ __align__(16) float lds_noise[2][2][NTHREADS][4];   // 16 KB
    __shared__ float lds_out[NTHREADS * 9];                        // 9 KB

    const int tid = threadIdx.x;
    const int b   = blockIdx.x * NTHREADS + tid;
    const int bc  = (b < n) ? b : (n - 1);    // clamp: every lane stays live
                                              // (barriers + EXEC-wide async ops)

    // Per-lane LDS byte offsets of the two 16B noise slots in each buffer.
    const uint32_t offA0 = (uint32_t)(uintptr_t)&lds_noise[0][0][tid][0];
    const uint32_t offB0 = (uint32_t)(uintptr_t)&lds_noise[0][1][tid][0];
    const uint32_t offA1 = (uint32_t)(uintptr_t)&lds_noise[1][0][tid][0];
    const uint32_t offB1 = (uint32_t)(uintptr_t)&lds_noise[1][1][tid][0];

    // ---- per-agent scalars -------------------------------------------------
    const float fx = fish_x[bc],  fy = fish_y[bc];
    const float px = prev_x[bc],  py = prev_y[bc];
    const float sp = speed[bc],   h  = heading[bc];
    const float ph = prev_heading[bc];
    const float ps = pred_speed[bc];
    const float pd = pred_hd[bc];

    const float dx = fx - px, dy = fy - py;
    const float actual_speed = __builtin_amdgcn_sqrtf(dx * dx + dy * dy);

    // atan2(sin(hd),cos(hd)) == wrap hd to [-pi,pi]: hd - 2pi*rint(hd/2pi)
    const float hd = h - ph;
    const float heading_delta =
        __builtin_fmaf(-6.283185307179586f,
                       __builtin_rintf(hd * 0.15915494309189535f), hd);

    const float wall = fmaxf(
        fmaxf(fmaxf((50.0f - fx) * 0.02f, 0.0f), fmaxf((fx - 750.0f) * 0.02f, 0.0f)),
        fmaxf(fmaxf((50.0f - fy) * 0.02f, 0.0f), fmaxf((fy - 550.0f) * 0.02f, 0.0f)));

    const float coll = ((actual_speed < 1.0f) && (sp > 0.5f)) ? 1.0f : 0.0f;

    const float ch = __cosf(h), sh = __sinf(h);   // v_cos_f32 / v_sin_f32

    // Fold I_TONIC (-1) and the +140 of the Izhikevich update into I:
    // base_i = I_i + 139 + 0.3*eps - u
    v2f Ic[NN / 2];
    Ic[0] = (v2f){ actual_speed * 5.0f + 139.0f,
                   fmaxf(0.0f, 3.0f - actual_speed) * 3.0f + 139.0f };
    Ic[1] = (v2f){ (ch + 1.0f) * 3.0f + 139.0f, (sh + 1.0f) * 3.0f + 139.0f };
    Ic[2] = (v2f){ wall * 12.0f + 139.0f,       wall * 12.0f + 139.0f };
    Ic[3] = (v2f){ coll * 15.0f + 139.0f,       coll * 15.0f + 139.0f };

    // ---- neuron state (32B-aligned vector loads, coalesced per wave) -------
    v2f v[NN / 2], u[NN / 2], rate[NN / 2];
    {
        const float4* vp = (const float4*)(v0 + (size_t)bc * NN);
        const float4* up = (const float4*)(u0 + (size_t)bc * NN);
        const float4 va = vp[0], vb = vp[1];
        const float4 ua = up[0], ub = up[1];
        v[0] = (v2f){va.x, va.y}; v[1] = (v2f){va.z, va.w};
        v[2] = (v2f){vb.x, vb.y}; v[3] = (v2f){vb.z, vb.w};
        u[0] = (v2f){ua.x, ua.y}; u[1] = (v2f){ua.z, ua.w};
        u[2] = (v2f){ub.x, ub.y}; u[3] = (v2f){ub.z, ub.w};
    }
    #pragma unroll
    for (int p = 0; p < NN / 2; ++p) rate[p] = (v2f){0.0f, 0.0f};

    const v2f c03   = {0.3f,   0.3f};
    const v2f c004  = {0.04f,  0.04f};
    const v2f c6    = {6.0f,   6.0f};
    const v2f c0004 = {0.004f, 0.004f};

    // ---- async double-buffered noise stream --------------------------------
    {   // prefetch step 0 -> buffer 0
        const float* g = noise + (size_t)bc * NN;
        async_ld16(offA0, g);
        async_ld16(offB0, g + 4);
    }

    #pragma unroll
    for (int s = 0; s < N_STEPS; ++s) {
        const int cur = s & 1;
        if (s + 1 < N_STEPS) {
            const float* g = noise + ((size_t)(s + 1) * n + bc) * NN;
            if (cur == 0) { async_ld16(offA1, g); async_ld16(offB1, g + 4); }
            else          { async_ld16(offA0, g); async_ld16(offB0, g + 4); }
            wait_async_le2();   // oldest 2 (step s) done; next 2 in flight
        } else {
            wait_async_0();
        }

        const float4 nA = *(const float4*)&lds_noise[cur][0][tid][0];
        const float4 nB = *(const float4*)&lds_noise[cur][1][tid][0];
        const v2f eps[NN / 2] = { (v2f){nA.x, nA.y}, (v2f){nA.z, nA.w},
                                  (v2f){nB.x, nB.y}, (v2f){nB.z, nB.w} };

        #pragma unroll
        for (int p = 0; p < NN / 2; ++p) {
            // packed-fp32 friendly arithmetic (v_pk_fma_f32 / v_pk_mul_f32)
            const v2f base = __builtin_elementwise_fma(eps[p], c03, Ic[p]) - u[p];
            const v2f t    = __builtin_elementwise_fma(v[p], c004, c6);
            const v2f vn   = __builtin_elementwise_fma(v[p], t, base);
            const v2f un   = __builtin_elementwise_fma(vn, c0004, u[p] * 0.98f);
            const v2f u8   = un + 8.0f;
            const v2f r9   = rate[p] * 0.9f;
            const v2f r1   = r9 + 0.1f;

            // scalar spike logic: one v_cmp + shared-VCC cndmasks per element
            const bool s0 = vn.x >= 30.0f;
            const bool s1 = vn.y >= 30.0f;
            v[p].x    = s0 ? -65.0f : vn.x;  v[p].y    = s1 ? -65.0f : vn.y;
            u[p].x    = s0 ? u8.x   : un.x;  u[p].y    = s1 ? u8.y   : un.y;
            rate[p].x = s0 ? r1.x   : r9.x;  rate[p].y = s1 ? r1.y   : r9.y;
        }
    }

    const v2f rsum2 = (rate[0] + rate[1]) + (rate[2] + rate[3]);
    const float rate_mean = (rsum2.x + rsum2.y) * 0.125f;

    // ---- precision-weighted prediction errors ------------------------------
    const float norm_speed = fminf(1.0f, actual_speed * 0.25f);
    const float e0 = norm_speed    - ps;
    const float e1 = heading_delta - pd;
    const float e2 = wall;                 // prediction ch2 = 0
    const float e3 = coll;                 // prediction ch3 = 0
    const float pe0 = e0, pe1 = e1, pe2 = 0.5f * e2, pe3 = e3;
    const float fe  = 0.5f * (e0 * e0 + e1 * e1 + 0.5f * e2 * e2 + e3 * e3);

    // ---- LDS transpose so the stride-9 output becomes coalesced ------------
    {
        const float res[9] = { actual_speed, heading_delta, wall, rate_mean,
                               pe0, pe1, pe2, pe3, fe };
        #pragma unroll
        for (int j = 0; j < 9; ++j)        // stride-9 (odd) -> bank-conflict-free
            lds_out[tid * 9 + j] = res[j];
    }
    __syncthreads();

    const size_t base  = (size_t)blockIdx.x * NTHREADS * 9;
    const size_t limit = (size_t)n * 9;
    #pragma unroll
    for (int k = 0; k < 9; ++k) {
        const size_t idx = (size_t)k * NTHREADS + tid;
        const size_t gi  = base + idx;
        if (gi < limit) out[gi] = lds_out[idx];
    }
}

extern "C" void kernel_launch(void* const* d_in, const int* in_sizes, int n_in,
                              void* d_out, int out_size, void* d_ws, size_t ws_size,
                              hipStream_t stream) {
    (void)n_in; (void)out_size; (void)d_ws; (void)ws_size;
    const int n = in_sizes[0];   // B
    const dim3 grid((n + NTHREADS - 1) / NTHREADS);
    fish_izhikevich_kernel<<<grid, NTHREADS, 0, stream>>>(
        (const float*)d_in[0],  (const float*)d_in[1],  (const float*)d_in[2],
        (const float*)d_in[3],  (const float*)d_in[4],  (const float*)d_in[5],
        (const float*)d_in[6],  (const float*)d_in[7],  (const float*)d_in[8],
        (const float*)d_in[9],  (const float*)d_in[10], (const float*)d_in[11],
        (float*)d_out, n);
}